// NeRFRenderer_12120397710038
// MI455X (gfx1250) — compile-verified
//
#include <hip/hip_runtime.h>
#include <stdint.h>

// NeRF importance sampling (near/far slab + inverse-CDF sample_pdf).
// Memory-bound (~137 MB traffic -> ~6us at 23.3 TB/s). One wave32 per ray.
// CDNA5 path: async global->LDS double-buffering of the 512B weight rows
// (global_load_async_to_lds_b128 + s_wait_asynccnt), wave-uniform slab math
// (scalarized by the compiler), shfl-based wave scan, LDS binary search.

#define NC 128              // coarse intervals (weights per ray)
#define NS 128              // n_importance samples per ray
#define WAVES 8             // waves (rays in flight) per block
#define BLOCK (WAVES * 32)
#define MAX_BLOCKS 2048

__global__ __launch_bounds__(BLOCK) void nerf_sample_pdf_kernel(
    const float* __restrict__ rays_o,
    const float* __restrict__ rays_d,
    const float* __restrict__ weights,
    float* __restrict__ out,
    int nrays)
{
    __shared__ __align__(16) float wbuf[WAVES][2][NC];   // async-staged weights (double buffered)
    __shared__ __align__(16) float cdfb[WAVES][NC];      // inclusive, normalized CDF (cdf[0]=0 implicit)

    const int lane = threadIdx.x & 31;
    const int wid  = threadIdx.x >> 5;
    const int gw   = blockIdx.x * WAVES + wid;       // global wave id == first ray id
    const int W    = gridDim.x * WAVES;              // ray stride

#if defined(__HIP_DEVICE_COMPILE__)
    // ---- initial async prefetch: 32 lanes x 16B = this ray's 128 weights ----
    if (gw < nrays) {
        unsigned long long g =
            (unsigned long long)(uintptr_t)(weights + (size_t)gw * NC + lane * 4);
        uint32_t loff = (uint32_t)(uintptr_t)&wbuf[wid][0][lane * 4];
        asm volatile("global_load_async_to_lds_b128 %0, %1, off"
                     :: "v"(loff), "v"(g) : "memory");
    }
#endif

    int buf = 0;
    for (int r = gw; r < nrays; r += W, buf ^= 1) {
#if defined(__HIP_DEVICE_COMPILE__)
        // ---- prefetch next ray into the other buffer, then wait for current ----
        int rn = r + W;
        if (rn < nrays) {
            unsigned long long g =
                (unsigned long long)(uintptr_t)(weights + (size_t)rn * NC + lane * 4);
            uint32_t loff = (uint32_t)(uintptr_t)&wbuf[wid][buf ^ 1][lane * 4];
            asm volatile("global_load_async_to_lds_b128 %0, %1, off"
                         :: "v"(loff), "v"(g) : "memory");
            // async loads complete in order: <=1 outstanding means current buf is ready
            asm volatile("s_wait_asynccnt 0x1" ::: "memory");
        } else {
            asm volatile("s_wait_asynccnt 0x0" ::: "memory");
        }
#endif

        // ---- near/far slab test (wave-uniform -> scalarized) ----
        const float ox = rays_o[3 * (size_t)r + 0];
        const float oy = rays_o[3 * (size_t)r + 1];
        const float oz = rays_o[3 * (size_t)r + 2];
        const float dx = rays_d[3 * (size_t)r + 0];
        const float dy = rays_d[3 * (size_t)r + 1];
        const float dz = rays_d[3 * (size_t)r + 2];

        const float ex = dx + 1e-15f, ey = dy + 1e-15f, ez = dz + 1e-15f;
        const float ax0 = (-2.0f - ox) / ex, ax1 = (2.0f - ox) / ex;
        const float ay0 = (-2.0f - oy) / ey, ay1 = (2.0f - oy) / ey;
        const float az0 = (-2.0f - oz) / ez, az1 = (2.0f - oz) / ez;

        float near = fmaxf(fmaxf(fminf(ax0, ax1), fminf(ay0, ay1)), fminf(az0, az1));
        float far  = fminf(fminf(fmaxf(ax0, ax1), fmaxf(ay0, ay1)), fmaxf(az0, az1));
        if (far < near) { near = 1e9f; far = 1e9f; }
        near = fmaxf(near, 0.05f);

        // ---- weights (from LDS) -> normalized inclusive CDF ----
        const float4 w4 = *(const float4*)&wbuf[wid][buf][lane * 4];
        const float w0 = w4.x + 1e-5f;
        const float w1 = w4.y + 1e-5f;
        const float w2 = w4.z + 1e-5f;
        const float w3 = w4.w + 1e-5f;
        const float p1 = w0;
        const float p2 = p1 + w1;
        const float p3 = p2 + w2;
        const float p4 = p3 + w3;

        float inc = p4;                       // wave32 inclusive scan of per-lane 4-sums
        #pragma unroll
        for (int off = 1; off < 32; off <<= 1) {
            float n = __shfl_up(inc, off, 32);
            if (lane >= off) inc += n;
        }
        const float total = __shfl(inc, 31, 32);
        const float base  = inc - p4;         // exclusive prefix for this lane
        const float rt    = 1.0f / total;

        float4 c4;
        c4.x = (base + p1) * rt;
        c4.y = (base + p2) * rt;
        c4.z = (base + p3) * rt;
        c4.w = (base + p4) * rt;
        *(float4*)&cdfb[wid][lane * 4] = c4;  // wave-private region; LDS in-order per wave

        // ---- 128 stratified samples: binary search + inverse-CDF lerp ----
        const float* cw = cdfb[wid];
        const float scale = (far - near) * (1.0f / 128.0f);
        float res[4];
        #pragma unroll
        for (int q = 0; q < 4; ++q) {
            const int   i = 4 * lane + q;
            const float u = (float)(2 * i + 1) * (1.0f / 256.0f);   // exact linspace value

            // upper_bound over CDF(k), k in [0,128], CDF(0)=0, CDF(k)=cw[k-1]
            int lo = 0, hi = NC + 1;
            #pragma unroll
            for (int it = 0; it < 8; ++it) {
                if (lo < hi) {
                    const int   mid = (lo + hi) >> 1;
                    const float cm  = (mid == 0) ? 0.0f : cw[mid - 1];
                    if (cm <= u) lo = mid + 1; else hi = mid;
                }
            }
            int below = lo - 1; if (below < 0)  below = 0;
            int above = lo;     if (above > NC) above = NC;

            const float cb = (below == 0) ? 0.0f : cw[below - 1];
            const float ca = (above == 0) ? 0.0f : cw[above - 1];
            float denom = ca - cb;
            if (denom < 1e-5f) denom = 1.0f;
            const float tt = (u - cb) / denom;
            // bins_k = near + (far-near)*k/128  (1/128 exact)
            res[q] = near + scale * ((float)below + tt * (float)(above - below));
        }
        float4 o4; o4.x = res[0]; o4.y = res[1]; o4.z = res[2]; o4.w = res[3];
        *(float4*)&out[(size_t)r * NS + 4 * lane] = o4;
    }
}

extern "C" void kernel_launch(void* const* d_in, const int* in_sizes, int n_in,
                              void* d_out, int out_size, void* d_ws, size_t ws_size,
                              hipStream_t stream) {
    const float* rays_o  = (const float*)d_in[0];
    const float* rays_d  = (const float*)d_in[1];
    const float* weights = (const float*)d_in[2];
    // d_in[3] is n_importance on-device; reference fixes it at 128 (compile-time NS).
    float* out = (float*)d_out;

    const int nrays = in_sizes[0] / 3;
    int blocks = (nrays + WAVES - 1) / WAVES;
    if (blocks > MAX_BLOCKS) blocks = MAX_BLOCKS;

    nerf_sample_pdf_kernel<<<blocks, BLOCK, 0, stream>>>(rays_o, rays_d, weights, out, nrays);
}